// TLSSM_68942815035606
// MI455X (gfx1250) — compile-verified
//
#include <hip/hip_runtime.h>
#include <hip/hip_bf16.h>

typedef __attribute__((ext_vector_type(2))) float v2f;
typedef __attribute__((ext_vector_type(4))) float v4f;
typedef __attribute__((ext_vector_type(8))) float v8f;

#define BATCH     256
#define D_MODEL   2048
#define D_SSM     4096
#define D_STATE   128
#define NHEADS    64
#define HEADDIM   64
#define CONV_DIM  4352            // D_SSM + 2*D_STATE
#define D_IN_PROJ 8512            // 2*D_SSM + 2*D_STATE + NHEADS

static __device__ __forceinline__ v8f wmma_f32(v2f a, v2f b, v8f c) {
    return __builtin_amdgcn_wmma_f32_16x16x4_f32(false, a, false, b, (short)0, c,
                                                 false, false);
}

// ---------------------------------------------------------------------------
// C[M,N] = X[M,K] @ W[N,K]^T  using V_WMMA_F32_16X16X4_F32.
// Wave tile: 32(M) x 64(N), 8 accumulators; 6 b64 loads feed 8 WMMAs.
// Register double-buffering: iteration k's WMMAs consume fragments loaded in
// iteration k-1 while k's loads are in flight (partial loadcnt waits).
// Block: 64 threads = 2 waves stacked along M -> block tile 64(M) x 64(N).
// f32 16x16x4 fragment layouts (wave32):
//   A (16x4):  lane L, reg v -> (M = L%16, K = 2*(L/16)+v)   => float2 load
//   B (4x16):  lane L, reg v -> (K = 2*(L/16)+v, N = L%16)   => float2 load
//   C/D:       lane L, reg r -> (M = r + 8*(L/16), N = L%16)
// ---------------------------------------------------------------------------
__global__ __launch_bounds__(64) void mamba_gemm_xwt_f32(
    const float* __restrict__ X, const float* __restrict__ W,
    float* __restrict__ C, int M, int N, int K)
{
    const int lane = threadIdx.x & 31;
    const int wave = threadIdx.x >> 5;
    const int m0 = blockIdx.y * 64 + wave * 32;
    const int n0 = blockIdx.x * 64;
    const int lm = lane & 15;
    const int lk = (lane >> 4) * 2;

    const float* ap0 = X + (size_t)(m0 + lm) * K + lk;
    const float* ap1 = ap0 + (size_t)16 * K;
    const float* bp0 = W + (size_t)(n0 + lm) * K + lk;
    const float* bp1 = bp0 + (size_t)16 * K;
    const float* bp2 = bp0 + (size_t)32 * K;
    const float* bp3 = bp0 + (size_t)48 * K;

    v8f a00 = {}, a01 = {}, a02 = {}, a03 = {};
    v8f a10 = {}, a11 = {}, a12 = {}, a13 = {};

    // prologue: fragments for k = 0
    v2f ca0 = *(const v2f*)(ap0);
    v2f ca1 = *(const v2f*)(ap1);
    v2f cb0 = *(const v2f*)(bp0);
    v2f cb1 = *(const v2f*)(bp1);
    v2f cb2 = *(const v2f*)(bp2);
    v2f cb3 = *(const v2f*)(bp3);

    #pragma unroll 4
    for (int k = 4; k < K; k += 4) {
        // prefetch next K-step (independent of the WMMAs below)
        v2f na0 = *(const v2f*)(ap0 + k);
        v2f na1 = *(const v2f*)(ap1 + k);
        v2f nb0 = *(const v2f*)(bp0 + k);
        v2f nb1 = *(const v2f*)(bp1 + k);
        v2f nb2 = *(const v2f*)(bp2 + k);
        v2f nb3 = *(const v2f*)(bp3 + k);

        a00 = wmma_f32(ca0, cb0, a00);
        a01 = wmma_f32(ca0, cb1, a01);
        a02 = wmma_f32(ca0, cb2, a02);
        a03 = wmma_f32(ca0, cb3, a03);
        a10 = wmma_f32(ca1, cb0, a10);
        a11 = wmma_f32(ca1, cb1, a11);
        a12 = wmma_f32(ca1, cb2, a12);
        a13 = wmma_f32(ca1, cb3, a13);

        ca0 = na0; ca1 = na1;
        cb0 = nb0; cb1 = nb1; cb2 = nb2; cb3 = nb3;
    }

    // epilogue K-step
    a00 = wmma_f32(ca0, cb0, a00);
    a01 = wmma_f32(ca0, cb1, a01);
    a02 = wmma_f32(ca0, cb2, a02);
    a03 = wmma_f32(ca0, cb3, a03);
    a10 = wmma_f32(ca1, cb0, a10);
    a11 = wmma_f32(ca1, cb1, a11);
    a12 = wmma_f32(ca1, cb2, a12);
    a13 = wmma_f32(ca1, cb3, a13);

    const int rbase = 8 * (lane >> 4);
    #pragma unroll
    for (int r = 0; r < 8; ++r) {
        size_t r0 = (size_t)(m0 + rbase + r) * N + n0 + lm;
        C[r0 +  0] = a00[r];
        C[r0 + 16] = a01[r];
        C[r0 + 32] = a02[r];
        C[r0 + 48] = a03[r];
        size_t r1 = (size_t)(m0 + 16 + rbase + r) * N + n0 + lm;
        C[r1 +  0] = a10[r];
        C[r1 + 16] = a11[r];
        C[r1 + 32] = a12[r];
        C[r1 + 48] = a13[r];
    }
}

// ---------------------------------------------------------------------------
// Depthwise conv state roll + conv + SiLU. One thread per (b, c).
// conv_state is touched exactly once -> non-temporal (keep L2 for weights).
// ---------------------------------------------------------------------------
__global__ __launch_bounds__(256) void mamba_conv_step(
    const float* __restrict__ zxbcdt, const float* __restrict__ conv_in,
    const float* __restrict__ conv_w, const float* __restrict__ conv_b,
    float* __restrict__ conv_out, float* __restrict__ xBC_out)
{
    int idx = blockIdx.x * 256 + threadIdx.x;        // b*CONV_DIM + c
    if (idx >= BATCH * CONV_DIM) return;
    int b = idx / CONV_DIM;
    int c = idx - b * CONV_DIM;

    v4f s = __builtin_nontemporal_load((const v4f*)conv_in + idx);
    float xnew = zxbcdt[(size_t)b * D_IN_PROJ + D_SSM + c];
    v4f w = ((const v4f*)conv_w)[c];

    v4f ns = { s.y, s.z, s.w, xnew };
    __builtin_nontemporal_store(ns, (v4f*)conv_out + idx);

    float acc = ns.x * w.x + ns.y * w.y + ns.z * w.z + ns.w * w.w + conv_b[c];
    float sig = 1.0f / (1.0f + __expf(-acc));
    xBC_out[idx] = acc * sig;
}

// ---------------------------------------------------------------------------
// SSM state update + y + SiLU gate. One block per (h, b): 64*128 = 8192 state
// elems streamed as v4f with non-temporal hints (1.07 GB stream-once; do not
// evict L2-resident projection weights). tid -> p = tid/4, q = tid%4.
// Reduction over n via wave32 shfl_xor (4 adjacent lanes share a p).
// ---------------------------------------------------------------------------
__global__ __launch_bounds__(256) void mamba_ssm_step(
    const float* __restrict__ zxbcdt, const float* __restrict__ xBC,
    const float* __restrict__ ssm_in, const float* __restrict__ dt_bias,
    const float* __restrict__ A_log, const float* __restrict__ Dvec,
    float* __restrict__ ssm_out, float* __restrict__ y_out)
{
    const int h = blockIdx.x;
    const int b = blockIdx.y;
    const int tid = threadIdx.x;

    __shared__ float sB[D_STATE];
    __shared__ float sC[D_STATE];

    const float* row = xBC + (size_t)b * CONV_DIM;
    if (tid < D_STATE)        sB[tid]           = row[D_SSM + tid];
    else if (tid < 2*D_STATE) sC[tid - D_STATE] = row[D_SSM + D_STATE + (tid - D_STATE)];

    float t  = zxbcdt[(size_t)b * D_IN_PROJ + 2 * D_SSM + 2 * D_STATE + h] + dt_bias[h];
    float dt = (t > 20.0f) ? t : log1pf(__expf(t));
    float dA = __expf(dt * (-__expf(A_log[h])));

    __syncthreads();

    const int p = tid >> 2;
    const int q = tid & 3;
    const float xp  = row[h * HEADDIM + p];
    const float dtx = dt * xp;

    const size_t base = ((((size_t)b * NHEADS + h) * HEADDIM) + p) * D_STATE + q * 32;
    const v4f* sin  = (const v4f*)(ssm_in  + base);
    v4f*       sout = (v4f*)      (ssm_out + base);

    float partial = 0.0f;
    #pragma unroll
    for (int j = 0; j < 8; ++j) {
        int n = q * 32 + j * 4;
        v4f s  = __builtin_nontemporal_load(sin + j);
        v4f Bv = *(const v4f*)(sB + n);
        v4f Cv = *(const v4f*)(sC + n);
        v4f ns = s * dA + dtx * Bv;
        __builtin_nontemporal_store(ns, sout + j);
        partial += ns.x * Cv.x + ns.y * Cv.y + ns.z * Cv.z + ns.w * Cv.w;
    }

    partial += __shfl_xor(partial, 1, 32);
    partial += __shfl_xor(partial, 2, 32);

    if (q == 0) {
        float y = partial + Dvec[h] * xp;
        float z = zxbcdt[(size_t)b * D_IN_PROJ + h * HEADDIM + p];
        float g = z / (1.0f + __expf(-z));
        y_out[(size_t)b * D_SSM + h * HEADDIM + p] = y * g;
    }
}

extern "C" void kernel_launch(void* const* d_in, const int* in_sizes, int n_in,
                              void* d_out, int out_size, void* d_ws, size_t ws_size,
                              hipStream_t stream) {
    const float* hs        = (const float*)d_in[0];
    const float* conv_in   = (const float*)d_in[1];
    const float* ssm_in    = (const float*)d_in[2];
    const float* in_proj_w = (const float*)d_in[3];
    const float* conv_w    = (const float*)d_in[4];
    const float* conv_b    = (const float*)d_in[5];
    const float* dt_bias   = (const float*)d_in[6];
    const float* A_log     = (const float*)d_in[7];
    const float* Dvec      = (const float*)d_in[8];
    const float* out_proj_w= (const float*)d_in[9];

    float* out      = (float*)d_out;                           // 256*2048
    float* conv_out = out + (size_t)BATCH * D_MODEL;           // 256*4352*4
    float* ssm_out  = conv_out + (size_t)BATCH * CONV_DIM * 4; // 256*64*64*128

    float* ws0 = (float*)d_ws;                                 // zxbcdt: 256*8512
    float* ws1 = ws0 + (size_t)BATCH * D_IN_PROJ;              // xBC:    256*4352
    float* ws2 = ws1 + (size_t)BATCH * CONV_DIM;               // y_gated:256*4096

    // 1) zxbcdt = hidden @ in_proj_w^T   (M=256, N=8512, K=2048)
    mamba_gemm_xwt_f32<<<dim3(D_IN_PROJ / 64, BATCH / 64), 64, 0, stream>>>(
        hs, in_proj_w, ws0, BATCH, D_IN_PROJ, D_MODEL);

    // 2) conv state roll + depthwise conv + SiLU
    mamba_conv_step<<<(BATCH * CONV_DIM + 255) / 256, 256, 0, stream>>>(
        ws0, conv_in, conv_w, conv_b, conv_out, ws1);

    // 3) SSM state update + gating
    mamba_ssm_step<<<dim3(NHEADS, BATCH), 256, 0, stream>>>(
        ws0, ws1, ssm_in, dt_bias, A_log, Dvec, ssm_out, ws2);

    // 4) out = y_gated @ out_proj_w^T   (M=256, N=2048, K=4096)
    mamba_gemm_xwt_f32<<<dim3(D_MODEL / 64, BATCH / 64), 64, 0, stream>>>(
        ws2, out_proj_w, out, BATCH, D_MODEL, D_SSM);
}